// CGBN_cuda_37984690766238
// MI455X (gfx1250) — compile-verified
//
#include <hip/hip_runtime.h>

// ---------------------------------------------------------------------------
// CG tensor-product layer (LMAX=5, tau=16 -> tau_out=32, B=256), fused:
//   FF on-the-fly from LDS-staged fragments  ->  V_WMMA_F32_16X16X4_F32 mixing
// A-tile complex values are computed once per (z,k) in "compute space" and
// redistributed to the WMMA A layout via wave shuffles (ds_bpermute).
// ---------------------------------------------------------------------------

typedef __attribute__((ext_vector_type(2))) float v2f;
typedef __attribute__((ext_vector_type(8))) float v8f;

#define LMAXP1 6
#define NTRIP 69
#define CG_TOTAL 24367
#define SH_STRIDE 577   // 576 float2 per z, +1 pad (1154 words -> bank offset 2/z)

// Triple tables in exact reference TRIPLES order (l-major, then l1, then l2)
__constant__ int c_TL[NTRIP] = {
  0,0,0,0,0,0,
  1,1,1,1,1,1,1,1,1,1,
  2,2,2,2,2,2,2,2,2,2,2,2,2,
  3,3,3,3,3,3,3,3,3,3,3,3,3,3,
  4,4,4,4,4,4,4,4,4,4,4,4,4,4,
  5,5,5,5,5,5,5,5,5,5,5,5};
__constant__ int c_TL1[NTRIP] = {
  0,1,2,3,4,5,
  0,1,1,2,2,3,3,4,4,5,
  0,1,1,1,2,2,2,3,3,3,4,4,5,
  0,1,1,1,2,2,2,2,3,3,3,4,4,5,
  0,1,1,1,2,2,2,2,3,3,3,4,4,5,
  0,1,1,2,2,2,3,3,3,4,4,5};
__constant__ int c_TL2[NTRIP] = {
  0,1,2,3,4,5,
  1,1,2,2,3,3,4,4,5,5,
  2,1,2,3,2,3,4,3,4,5,4,5,5,
  3,2,3,4,2,3,4,5,3,4,5,4,5,5,
  4,3,4,5,2,3,4,5,3,4,5,4,5,5,
  5,4,5,3,4,5,3,4,5,4,5,5};

__constant__ int c_tripleStart[LMAXP1] = {0, 6, 16, 29, 43, 57};
__constant__ int c_K[LMAXP1]       = {1536, 2560, 3328, 3584, 3584, 3072}; // T_FF[l]
__constant__ int c_cumW[LMAXP1]    = {0, 49152, 131072, 237568, 352256, 466944};
__constant__ int c_outBase[LMAXP1] = {0, 32, 128, 288, 512, 800};
// blocks: per l, 32 z-tiles * ceil((2l+1)/2) m-groups
__constant__ int c_blkBase[LMAXP1 + 1] = {0, 32, 96, 192, 320, 480, 672};

__device__ float g_cg[CG_TOTAL];

// ---------------------------------------------------------------------------
// Kernel 0: fill CG coefficient blocks (Racah formula), recomputed per launch
// ---------------------------------------------------------------------------
__device__ inline double dfact(int n) {
  double r = 1.0;
  for (int i = 2; i <= n; ++i) r *= (double)i;
  return r;
}

__device__ double cg_coef(int l1, int m1, int l2, int m2, int l, int m) {
  if (m1 + m2 != m) return 0.0;
  double pre = sqrt((double)(2 * l + 1) * dfact(l + l1 - l2) * dfact(l - l1 + l2) *
                    dfact(l1 + l2 - l) / dfact(l1 + l2 + l + 1));
  pre *= sqrt(dfact(l + m) * dfact(l - m) * dfact(l1 - m1) * dfact(l1 + m1) *
              dfact(l2 - m2) * dfact(l2 + m2));
  int kmin = max(0, max(l2 - l - m1, l1 + m2 - l));
  int kmax = min(l1 + l2 - l, min(l1 - m1, l2 + m2));
  double s = 0.0;
  for (int k = kmin; k <= kmax; ++k) {
    double den = dfact(k) * dfact(l1 + l2 - l - k) * dfact(l1 - m1 - k) *
                 dfact(l2 + m2 - k) * dfact(l - l2 + m1 + k) * dfact(l - l1 - m2 + k);
    s += ((k & 1) ? -1.0 : 1.0) / den;
  }
  return pre * s;
}

__global__ __launch_bounds__(64) void cg_init_kernel() {
  int j = blockIdx.x;                    // one block per triple
  int l = c_TL[j], l1 = c_TL1[j], l2 = c_TL2[j];
  int off = 0;
  for (int q = 0; q < j; ++q)
    off += (2 * c_TL[q] + 1) * (2 * c_TL1[q] + 1) * (2 * c_TL2[q] + 1);
  int n1 = 2 * l1 + 1, n2 = 2 * l2 + 1, nm = 2 * l + 1;
  int total = nm * n1 * n2;
  for (int e = threadIdx.x; e < total; e += blockDim.x) {
    int m2i = e % n2;
    int rest = e / n2;
    int m1i = rest % n1;
    int mi = rest / n1;
    int m = mi - l, m1 = m1i - l1, m2v = m2i - l2;
    float v = 0.0f;
    if (m1 + m2v == m) v = (float)cg_coef(l1, m1, l2, m2v, l, m);
    g_cg[off + e] = v;
  }
}

// ---------------------------------------------------------------------------
// Main fused kernel: block = (l, z-tile of 8, m-pair), 2 waves (one m each).
// Each wave: D tile 16x32, rows = {Re z0..z7, Im z0..z7}, K = T_FF[l] step 4.
// ---------------------------------------------------------------------------
__global__ __launch_bounds__(64) void cgbn_fused_kernel(const float2* __restrict__ Fs,
                                                        const float* __restrict__ W,
                                                        float* __restrict__ out) {
  __shared__ float2 shF[8 * SH_STRIDE];
  __shared__ int s_l1[16], s_l2[16], s_off[16];

  int bid = blockIdx.x;
  int l = 0;
  while (l < 5 && bid >= c_blkBase[l + 1]) ++l;
  int rem = bid - c_blkBase[l];
  int mg = rem >> 5;        // m-group
  int zt = rem & 31;        // z-tile
  int z0 = zt * 8;
  int tid = threadIdx.x;

  if (tid == 0) {
    int off = 0;
    int ts = c_tripleStart[l];
    for (int j = 0; j < NTRIP; ++j) {
      int sz = (2 * c_TL[j] + 1) * (2 * c_TL1[j] + 1) * (2 * c_TL2[j] + 1);
      if (c_TL[j] == l) {
        int jl = j - ts;
        s_l1[jl] = c_TL1[j];
        s_l2[jl] = c_TL2[j];
        s_off[jl] = off;
      }
      off += sz;
    }
  }
  // stage 8 batch rows of the full 576-complex activation vector
  for (int i = tid; i < 8 * 576; i += 64) {
    int zl = i / 576, idx = i - zl * 576;
    shF[zl * SH_STRIDE + idx] = Fs[(z0 + zl) * 576 + idx];
  }
  __syncthreads();

  int wave = tid >> 5;
  int lane = tid & 31;
  int m = mg * 2 + wave;          // mi index 0..2l
  if (m > 2 * l) return;          // odd (2l+1): second wave may idle

  int K = c_K[l];
  const float* Wl = W + c_cumW[l];

  // Compute-space decode: lane -> (z = lane&7, kslot = lane>>3), one complex
  // FF value per lane per K-step (each of the 8z x 4k values computed once).
  int zl    = lane & 7;
  int kslot = lane >> 3;          // 0..3
  int comp  = (lane >> 3) & 1;    // component of this lane's WMMA A row
  int hi    = lane >> 4;
  int klane = hi * 2;
  int n16   = lane & 15;          // B/D: N = lane%16 (+16 for second half)
  int srcA  = hi * 16 + zl;       // shuffle source for a.x (a.y = srcA+8)
  const float2* shz = &shF[zl * SH_STRIDE];

  v8f acc0 = {0.f, 0.f, 0.f, 0.f, 0.f, 0.f, 0.f, 0.f};
  v8f acc1 = {0.f, 0.f, 0.f, 0.f, 0.f, 0.f, 0.f, 0.f};

  for (int kb = 0; kb < K; kb += 4) {
    __builtin_prefetch(Wl + (kb + 16) * 32, 0, 3);
    // ---- one complex FF value per lane -------------------------------------
    int k = kb + kslot;            // k-steps of 4 never straddle a triple
    int j0 = k >> 8;               // local triple (uniform per wave)
    int t = (k >> 4) & 15;
    int s = k & 15;
    int l1 = s_l1[j0], l2 = s_l2[j0], off = s_off[j0];
    int d = m - l + l1 + l2;       // m2i = d - m1i
    int m1lo = max(0, d - 2 * l2);
    int m1hi = min(2 * l1, d);     // bounds wave-uniform
    const float2* f1 = shz + l1 * l1 * 16 + t;
    const float2* f2 = shz + l2 * l2 * 16 + s;
    int cgrow = off + (m * (2 * l1 + 1) + m1lo) * (2 * l2 + 1) + (d - m1lo);
    float re = 0.f, im = 0.f;
    for (int m1i = m1lo; m1i <= m1hi; ++m1i) {
      float c = g_cg[cgrow];
      cgrow += 2 * l2;             // +(2l2+1) in m1 dim, -1 in m2 dim
      float2 A1 = f1[m1i * 16];
      float2 A2 = f2[(d - m1i) * 16];
      re = fmaf(c, A1.x * A2.x - A1.y * A2.y, re);
      im = fmaf(c, A1.x * A2.y + A1.y * A2.x, im);
    }
    // ---- redistribute to WMMA A layout: lane -> (M=lane%16, K=2*(lane/16)+u)
    float rx = __shfl(re, srcA);
    float ix = __shfl(im, srcA);
    float ry = __shfl(re, srcA + 8);
    float iy = __shfl(im, srcA + 8);
    v2f a;
    a.x = comp ? ix : rx;
    a.y = comp ? iy : ry;
    // ---- B tiles: lane -> N = lane%16, K cols = 2*(lane/16)+{0,1} ----------
    int kb2 = kb + klane;
    v2f b0, b1;
    b0.x = Wl[kb2 * 32 + n16];
    b0.y = Wl[(kb2 + 1) * 32 + n16];
    b1.x = Wl[kb2 * 32 + 16 + n16];
    b1.y = Wl[(kb2 + 1) * 32 + 16 + n16];
    acc0 = __builtin_amdgcn_wmma_f32_16x16x4_f32(false, a, false, b0, (short)0, acc0,
                                                 false, false);
    acc1 = __builtin_amdgcn_wmma_f32_16x16x4_f32(false, a, false, b1, (short)0, acc1,
                                                 false, false);
  }

  // D layout: VGPR r -> M = r + 8*hi => comp = hi, z-local = r
  int obase = c_outBase[l] + m * 32;
#pragma unroll
  for (int r = 0; r < 8; ++r) {
    int z = z0 + r;
    size_t base = (size_t)z * 1152 + obase;
    out[(base + n16) * 2 + hi]      = acc0[r];
    out[(base + 16 + n16) * 2 + hi] = acc1[r];
  }
}

// ---------------------------------------------------------------------------
extern "C" void kernel_launch(void* const* d_in, const int* in_sizes, int n_in,
                              void* d_out, int out_size, void* d_ws, size_t ws_size,
                              hipStream_t stream) {
  (void)in_sizes; (void)n_in; (void)d_ws; (void)ws_size; (void)out_size;
  const float2* Fs = (const float2*)d_in[0];  // (256, 576, 2) f32 -> float2
  const float* W   = (const float*)d_in[1];   // (565248,) f32
  float* out       = (float*)d_out;           // (256, 1152, 2) f32

  cg_init_kernel<<<NTRIP, 64, 0, stream>>>();
  cgbn_fused_kernel<<<672, 64, 0, stream>>>(Fs, W, out);
}